// KGAT_3582002725212
// MI455X (gfx1250) — compile-verified
//
#include <hip/hip_runtime.h>

typedef __attribute__((ext_vector_type(2))) float v2f;
typedef __attribute__((ext_vector_type(8))) float v8f;

#define OUTD 176
#define EPS_L2 1e-12f

__device__ __forceinline__ float lrelu(float x) { return x > 0.f ? x : 0.01f * x; }

// ---------------- zero fill ----------------
__global__ void zero_kernel(float* __restrict__ p, long n) {
    long i = (long)blockIdx.x * blockDim.x + threadIdx.x;
    if (i < n) p[i] = 0.f;
}

// ---------------- copy layer-0 embedding into output cols [0,64) ----------------
__global__ void copy_embed_kernel(const float* __restrict__ src, float* __restrict__ out,
                                  int n, int d, int colOff) {
    long i = (long)blockIdx.x * blockDim.x + threadIdx.x;
    long total = (long)n * d;
    if (i >= total) return;
    int node = (int)(i / d);
    int c    = (int)(i % d);
    out[(long)node * OUTD + colOff + c] = src[i];
}

// ---------------- SpMM: side[row] += val * x[col]  (edge-parallel, L2 atomics) ----------------
__global__ void spmm_kernel(const float* __restrict__ val, const int* __restrict__ row,
                            const int* __restrict__ col, const float* __restrict__ x,
                            float* __restrict__ y, int E, int D) {
    int chunks = D >> 2;                       // float4 chunks per row
    long total = (long)E * chunks;
    long i = (long)blockIdx.x * blockDim.x + threadIdx.x;
    if (i >= total) return;
    int e = (int)(i / chunks);
    int c = ((int)(i % chunks)) << 2;
    float v  = val[e];
    long src = (long)col[e] * D + c;
    long dst = (long)row[e] * D + c;
    float4 xv = *(const float4*)(x + src);
    atomicAdd(y + dst + 0, v * xv.x);
    atomicAdd(y + dst + 1, v * xv.y);
    atomicAdd(y + dst + 2, v * xv.z);
    atomicAdd(y + dst + 3, v * xv.w);
}

// ---------------- Fused aggregator + L2 norm (weights staged in LDS) ----------------
// ego_next = lrelu((ego+side)W1^T+b1) + lrelu((ego*side)W2^T+b2)   [raw -> egoNext]
// out[:, colOff:colOff+dout] = l2norm(ego_next)                    [normalized -> out]
//
// One wave owns 16 nodes x full dout row (NT = dout/16 N-tiles); ego/side are
// fetched exactly once from global with immediate offsets, W1/W2 are staged to
// LDS once per block and B-fragments read via ds_load_b64 immediate offsets.
// V_WMMA_F32_16X16X4_F32 layouts:
//   A-frag (16x4, 2 VGPR): lane l holds X[l&15][k + 2*(l>>4) + {0,1}]
//   B-frag (4x16, 2 VGPR): lane l holds W[l&15][k + 2*(l>>4) + {0,1}]
//   D      (16x16, 8 VGPR): lane l, reg v -> M = 8*(l>>4)+v, N = l&15
template <int NT, int DIN>
__global__ __launch_bounds__(128)
void aggregate_norm_wmma_kernel(const float* __restrict__ ego, const float* __restrict__ side,
                                const float* __restrict__ w1, const float* __restrict__ b1,
                                const float* __restrict__ w2, const float* __restrict__ b2,
                                float* __restrict__ egoNext, float* __restrict__ out,
                                int n, int colOff) {
    constexpr int dout = NT * 16;
    constexpr int WELEMS = dout * DIN;

    __shared__ __align__(16) float sw1[WELEMS];
    __shared__ __align__(16) float sw2[WELEMS];

    // Cooperative stage of both weight matrices into LDS (float4 granularity).
    for (int i = threadIdx.x * 4; i < WELEMS; i += blockDim.x * 4) {
        *(float4*)(sw1 + i) = *(const float4*)(w1 + i);
        *(float4*)(sw2 + i) = *(const float4*)(w2 + i);
    }
    __syncthreads();

    int lane = threadIdx.x & 31;
    int wave = threadIdx.x >> 5;
    int m0   = (blockIdx.x * 4 + wave) * 16;   // 16-node tile
    if (m0 >= n) return;                        // uniform per wave: EXEC all-1s for WMMA

    int mrow  = lane & 15;
    int ncol  = lane & 15;
    int khalf = (lane >> 4) * 2;                // 0 or 2

    const float* egoRow  = ego  + (long)(m0 + mrow) * DIN + khalf;
    const float* sideRow = side + (long)(m0 + mrow) * DIN + khalf;
    const float* sb1 = sw1 + ncol * DIN + khalf;   // + (t*16*DIN + k) immediate
    const float* sb2 = sw2 + ncol * DIN + khalf;

    v8f acc1[NT], acc2[NT];
#pragma unroll
    for (int t = 0; t < NT; ++t) { acc1[t] = {}; acc2[t] = {}; }

#pragma unroll
    for (int k = 0; k < DIN; k += 4) {
        v2f e = *(const v2f*)(egoRow + k);
        v2f s = *(const v2f*)(sideRow + k);
        v2f aSum  = e + s;                      // v_pk_add_f32
        v2f aProd = e * s;                      // v_pk_mul_f32
#pragma unroll
        for (int t = 0; t < NT; ++t) {
            v2f bw1 = *(const v2f*)(sb1 + t * 16 * DIN + k);   // ds_load_b64 offset imm
            v2f bw2 = *(const v2f*)(sb2 + t * 16 * DIN + k);
            acc1[t] = __builtin_amdgcn_wmma_f32_16x16x4_f32(false, aSum,  false, bw1,
                                                            (short)0, acc1[t], false, false);
            acc2[t] = __builtin_amdgcn_wmma_f32_16x16x4_f32(false, aProd, false, bw2,
                                                            (short)0, acc2[t], false, false);
        }
    }

    // Epilogue: bias + lrelu + sum, accumulate row sum-of-squares (reuse acc1 for r).
    float ss[8];
#pragma unroll
    for (int v = 0; v < 8; ++v) ss[v] = 0.f;
#pragma unroll
    for (int t = 0; t < NT; ++t) {
        float bias1 = b1[t * 16 + ncol];
        float bias2 = b2[t * 16 + ncol];
#pragma unroll
        for (int v = 0; v < 8; ++v) {
            float r = lrelu(acc1[t][v] + bias1) + lrelu(acc2[t][v] + bias2);
            acc1[t][v] = r;
            ss[v] += r * r;
        }
    }
    // Reduce over the 16 N-lanes of each half-wave (masks 1,2,4,8 stay in-half).
#pragma unroll
    for (int v = 0; v < 8; ++v) {
        ss[v] += __shfl_xor(ss[v], 1);
        ss[v] += __shfl_xor(ss[v], 2);
        ss[v] += __shfl_xor(ss[v], 4);
        ss[v] += __shfl_xor(ss[v], 8);
    }

    int mbase = m0 + (lane >> 4) * 8;
#pragma unroll
    for (int v = 0; v < 8; ++v) {
        float scale = 1.f / fmaxf(sqrtf(ss[v]), EPS_L2);
        long rowE = (long)(mbase + v) * dout;
        long rowO = (long)(mbase + v) * OUTD + colOff;
#pragma unroll
        for (int t = 0; t < NT; ++t) {
            float r = acc1[t][v];
            egoNext[rowE + t * 16 + ncol] = r;            // raw ego for next layer
            out[rowO + t * 16 + ncol]     = r * scale;    // normalized slice
        }
    }
}

extern "C" void kernel_launch(void* const* d_in, const int* in_sizes, int n_in,
                              void* d_out, int out_size, void* d_ws, size_t ws_size,
                              hipStream_t stream) {
    const float* embed = (const float*)d_in[0];
    const float* w1_0 = (const float*)d_in[1];
    const float* b1_0 = (const float*)d_in[2];
    const float* w2_0 = (const float*)d_in[3];
    const float* b2_0 = (const float*)d_in[4];
    const float* w1_1 = (const float*)d_in[5];
    const float* b1_1 = (const float*)d_in[6];
    const float* w2_1 = (const float*)d_in[7];
    const float* b2_1 = (const float*)d_in[8];
    const float* w1_2 = (const float*)d_in[9];
    const float* b1_2 = (const float*)d_in[10];
    const float* w2_2 = (const float*)d_in[11];
    const float* b2_2 = (const float*)d_in[12];
    const float* edge_val = (const float*)d_in[13];
    const int*   edge_row = (const int*)d_in[14];
    const int*   edge_col = (const int*)d_in[15];

    const int nNodes = in_sizes[0] / 64;
    const int E      = in_sizes[13];
    float* out = (float*)d_out;

    // workspace: ego ping-pong + side buffer (each nNodes*64 floats)
    float* egoA = (float*)d_ws;
    float* egoB = egoA + (long)nNodes * 64;
    float* side = egoB + (long)nNodes * 64;

    int mtiles = (nNodes + 15) / 16;
    int gemmBlocks = (mtiles + 3) / 4;

    auto run_spmm = [&](const float* ego, int din) {
        long sideElems = (long)nNodes * din;
        zero_kernel<<<(int)((sideElems + 255) / 256), 256, 0, stream>>>(side, sideElems);
        long spmmThreads = (long)E * (din / 4);
        spmm_kernel<<<(int)((spmmThreads + 255) / 256), 256, 0, stream>>>(
            edge_val, edge_row, edge_col, ego, side, E, din);
    };

    // all_embed[0] = raw input embedding -> cols [0,64)
    {
        long total = (long)nNodes * 64;
        copy_embed_kernel<<<(int)((total + 255) / 256), 256, 0, stream>>>(embed, out, nNodes, 64, 0);
    }

    // Layer 0: din=64, dout=64 -> cols [64,128)
    run_spmm(embed, 64);
    aggregate_norm_wmma_kernel<4, 64><<<gemmBlocks, 128, 0, stream>>>(
        embed, side, w1_0, b1_0, w2_0, b2_0, egoA, out, nNodes, 64);

    // Layer 1: din=64, dout=32 -> cols [128,160)
    run_spmm(egoA, 64);
    aggregate_norm_wmma_kernel<2, 64><<<gemmBlocks, 128, 0, stream>>>(
        egoA, side, w1_1, b1_1, w2_1, b2_1, egoB, out, nNodes, 128);

    // Layer 2: din=32, dout=16 -> cols [160,176)
    run_spmm(egoB, 32);
    aggregate_norm_wmma_kernel<1, 32><<<gemmBlocks, 128, 0, stream>>>(
        egoB, side, w1_2, b1_2, w2_2, b2_2, egoA, out, nNodes, 160);
}